// AdapterAttnForMamba_2920577761339
// MI455X (gfx1250) — compile-verified
//
#include <hip/hip_runtime.h>

// AdapterAttnForMamba fused kernel for gfx1250 (MI455X).
// Depthwise 3x3 conv over pad-token-extended 66x66 image, crop, transpose, residual.
// Memory-bound (~540MB @ 23.3 TB/s => ~23us floor). Uses async global->LDS DMA
// (ASYNCcnt path) to stage conv tap rows, ds_load_b128 + v_fmac_f32 compute,
// and non-temporal b128 stores so the write stream doesn't evict x from L2.

#define BB 16
#define HH 64
#define WW 64
#define CC 1024
#define CHUNK 128                 // channels per block
#define NCHUNK (CC / CHUNK)       // 8
#define GRP (CHUNK / 4)           // 32 float4 channel-groups == wave32 lanes
#define THREADS 256               // 8 waves
#define ROWF (WW * CHUNK)         // floats per staged row in LDS (8192)

typedef __attribute__((ext_vector_type(4))) int v4i;
typedef __attribute__((ext_vector_type(4))) float v4f;
typedef __attribute__((address_space(1))) v4i gv4i;   // global
typedef __attribute__((address_space(3))) v4i sv4i;   // LDS

// ---- gfx1250 async global->LDS copy (16B per lane), tracked by ASYNCcnt ----
__device__ __forceinline__ void async_cp16(const float* __restrict__ g, float* l) {
#if __has_builtin(__builtin_amdgcn_global_load_async_to_lds_b128)
  __builtin_amdgcn_global_load_async_to_lds_b128((gv4i*)g, (sv4i*)l, 0, 0);
#else
  unsigned loff = (unsigned)(unsigned long long)(__attribute__((address_space(3))) char*)l;
  asm volatile("global_load_async_to_lds_b128 %0, %1, off"
               :: "v"(loff), "v"(g) : "memory");
#endif
}

__device__ __forceinline__ void wait_async0() {
#if __has_builtin(__builtin_amdgcn_s_wait_asynccnt)
  __builtin_amdgcn_s_wait_asynccnt(0);
#else
  asm volatile("s_wait_asynccnt 0" ::: "memory");
#endif
}

__device__ __forceinline__ void fma4(float4& a, const float4 w, const float4 v) {
  a.x = fmaf(w.x, v.x, a.x);
  a.y = fmaf(w.y, v.y, a.y);
  a.z = fmaf(w.z, v.z, a.z);
  a.w = fmaf(w.w, v.w, a.w);
}

#define W4(k) make_float4(wbuf[(k)], wbuf[9 + (k)], wbuf[18 + (k)], wbuf[27 + (k)])

__global__ __launch_bounds__(THREADS) void dwconv_pad_fused_kernel(
    const float* __restrict__ x, const float* __restrict__ pad_token,
    const float* __restrict__ conv_w, const float* __restrict__ conv_b,
    float* __restrict__ out) {
  // 3 tap rows x 64 cols x 128 channels, fp32 = 96KB (3 WGs / 320KB WGP)
  __shared__ float lds[3 * ROWF];

  const int tid = threadIdx.x;
  const int bid = blockIdx.x;
  const int chunk = bid & (NCHUNK - 1);
  const int i = (bid / NCHUNK) & (HH - 1);   // conv output row
  const int b = bid / (NCHUNK * HH);
  const int c0 = chunk * CHUNK;

  const size_t batch_base = (size_t)b * (HH * WW) * CC;

  // ---------- stage valid tap rows (i-1, i, i+1) into LDS via async DMA ----------
#pragma unroll
  for (int m = 0; m < 3; ++m) {
    const int r = i - 1 + m;
    if (r < 0 || r >= HH) continue;  // block-uniform
    const float* src_row = x + batch_base + (size_t)r * WW * CC + c0;
    float* dst_row = lds + m * ROWF;
#pragma unroll
    for (int q = 0; q < (WW * GRP) / THREADS; ++q) {  // 8 issues/row
      const int idx = q * THREADS + tid;
      const int col = idx >> 5;   // 0..63
      const int grp = idx & 31;   // channel float4 group
      async_cp16(src_row + (size_t)col * CC + grp * 4,
                 dst_row + col * CHUNK + grp * 4);
    }
  }

  // ---------- per-thread constants while DMA is in flight ----------
  const int lane = tid & 31;        // channel group (uniform j per wave below)
  const int wv = tid >> 5;          // wave id -> pixel base
  const int ch = c0 + lane * 4;

  alignas(16) float wbuf[36];       // 3x3 weights for 4 consecutive channels
  {
    const float4* wp = (const float4*)(conv_w + (size_t)ch * 9);  // 16B aligned (144B stride)
#pragma unroll
    for (int t = 0; t < 9; ++t) ((float4*)wbuf)[t] = wp[t];
  }
  const float4 bias4 = *(const float4*)(conv_b + ch);

  // pad_token layout [C][2]; deswizzle into per-parity float4s
  float4 pt[2];
  {
    const float4* pp = (const float4*)(pad_token + (size_t)ch * 2);
    const float4 A = pp[0];
    const float4 Bv = pp[1];
    pt[0] = make_float4(A.x, A.z, Bv.x, Bv.z);
    pt[1] = make_float4(A.y, A.w, Bv.y, Bv.w);
  }

  wait_async0();
  __syncthreads();

  // ---------- compute 8 output pixels per thread ----------
#pragma unroll
  for (int k = 0; k < 8; ++k) {
    const int j = wv + 8 * k;       // conv output col, uniform within a wave
    float4 acc = bias4;
#pragma unroll
    for (int m = 0; m < 3; ++m) {
      const int r = i - 1 + m;
      if (r < 0) continue;          // zero 'SAME' pad row (block-uniform)
      if (r >= HH) {
        // padded row 64: every tap is pad_token[col & 1] (col 64 -> parity 0)
#pragma unroll
        for (int dx = 0; dx < 3; ++dx) {
          const int c = j - 1 + dx;
          if (c >= 0) fma4(acc, W4(m * 3 + dx), pt[c & 1]);
        }
      } else {
#pragma unroll
        for (int dx = 0; dx < 3; ++dx) {
          const int c = j - 1 + dx;
          const int cl = c < 0 ? 0 : (c >= WW ? WW - 1 : c);  // clamp: safe LDS addr
          float4 v = *(const float4*)&lds[m * ROWF + cl * CHUNK + lane * 4];
          if (c >= WW) v = pt[0];                 // padded col 64 -> token parity 0
          float4 wk = W4(m * 3 + dx);
          if (c < 0) wk = make_float4(0.f, 0.f, 0.f, 0.f);  // zero 'SAME' pad col
          fma4(acc, wk, v);
        }
      }
    }
    // transpose gather baked in: out[b, j*64+i] = x[b, j*64+i] + y_conv(i, j)
    const size_t go = batch_base + (size_t)(j * HH + i) * CC + ch;
    const float4 xr = *(const float4*)(x + go);
    v4f o;
    o.x = xr.x + acc.x;
    o.y = xr.y + acc.y;
    o.z = xr.z + acc.z;
    o.w = xr.w + acc.w;
    // write-once stream: non-temporal hint keeps x rows resident in L2
    __builtin_nontemporal_store(o, (v4f*)(out + go));
  }
}

extern "C" void kernel_launch(void* const* d_in, const int* in_sizes, int n_in,
                              void* d_out, int out_size, void* d_ws, size_t ws_size,
                              hipStream_t stream) {
  const float* x = (const float*)d_in[0];
  const float* pad_token = (const float*)d_in[1];
  const float* conv_w = (const float*)d_in[2];
  const float* conv_b = (const float*)d_in[3];
  // d_in[4] = seq_idx (int32): pure h/w transpose, baked into the index math.
  float* out = (float*)d_out;

  dim3 grid(BB * HH * NCHUNK);   // 8192 blocks
  dwconv_pad_fused_kernel<<<grid, THREADS, 0, stream>>>(x, pad_token, conv_w, conv_b, out);
}